// L2loss_67327907332547
// MI455X (gfx1250) — compile-verified
//
#include <hip/hip_runtime.h>
#include <math.h>

#define H_PIX 50176          // 224*224
#define NBINS 8192
#define THREADS 256
#define EPT 32               // elements per thread in scan: 256*32 = 8192

typedef __attribute__((ext_vector_type(2))) float v2f;
typedef __attribute__((ext_vector_type(8))) float v8f;

// ---------------------------------------------------------------------------
// Kernel 1: per-channel cumsum -> int32 truncate -> clip to [0, H].
// 6 blocks: b in [0,3) -> target channel b ; b in [3,6) -> output channel b-3.
// 256-thread block scan; the 256 partial sums are prefix-scanned on the
// matrix pipe: T1 = M (16x16) * U (upper-triangular ones) via 4 chained
// v_wmma_f32_16x16x4_f32 (K=16 split into 4 chunks of K=4).
// ---------------------------------------------------------------------------
__global__ void scan_kernel(const float* __restrict__ target,
                            const float* __restrict__ output,
                            int* __restrict__ cint /* 6*NBINS ints */) {
  __shared__ float sums[THREADS];
  __shared__ float rowpref[256];   // T1 = row-inclusive prefixes, 16x16

  const int b   = blockIdx.x;      // 0..5
  const int ch  = (b < 3) ? b : (b - 3);
  const float* __restrict__ src = ((b < 3) ? target : output) + ch * NBINS;
  int* __restrict__ dst = cint + b * NBINS;
  const int tid = threadIdx.x;

  // Serial local sum of 32 contiguous elements.
  float x[EPT];
  float s = 0.0f;
#pragma unroll
  for (int e = 0; e < EPT; ++e) {
    x[e] = src[tid * EPT + e];
    s += x[e];
  }
  sums[tid] = s;
  __syncthreads();

  // Wave 0 (lanes 0..31, EXEC all ones): T1 = M * U on the WMMA pipe.
  if (tid < 32) {
    const int  l  = tid;
    const bool hi = (l >= 16);
    const int  r  = hi ? (l - 16) : l;   // A: M-row ; B/D: N-column
    v8f acc = {};
#pragma unroll
    for (int c = 0; c < 4; ++c) {
      // ISA A-layout (16x4 f32): lanes 0-15 hold K = 4c+{0,1}; lanes 16-31
      // hold K = 4c+{2,3}, M = lane%16.  B-layout (4x16): same K split,
      // N = lane%16.
      const int k0 = 4 * c + (hi ? 2 : 0);
      const int k1 = k0 + 1;
      v2f a, u;
      a.x = sums[r * 16 + k0];
      a.y = sums[r * 16 + k1];
      u.x = (k0 <= r) ? 1.0f : 0.0f;   // U[k][n] = (k <= n)
      u.y = (k1 <= r) ? 1.0f : 0.0f;
      acc = __builtin_amdgcn_wmma_f32_16x16x4_f32(
          /*neg_a=*/false, a, /*neg_b=*/false, u,
          /*c_mod=*/(short)0, acc, /*reuse_a=*/false, /*reuse_b=*/false);
    }
    // D layout: lanes 0-15 -> rows 0-7 in vgprs 0-7 at N=lane;
    //           lanes 16-31 -> rows 8-15 at N=lane-16.
#pragma unroll
    for (int v = 0; v < 8; ++v) {
      const int row = v + (hi ? 8 : 0);
      rowpref[row * 16 + r] = acc[v];
    }
  }
  __syncthreads();

  // Exclusive flat prefix for this thread = sum of full rows above
  // + inclusive row prefix at (i,j) - own partial.
  const int i = tid >> 4, j = tid & 15;
  float base = 0.0f;
  for (int ii = 0; ii < i; ++ii) base += rowpref[ii * 16 + 15];
  float run = base + rowpref[i * 16 + j] - sums[tid];

#pragma unroll
  for (int e = 0; e < EPT; ++e) {
    run += x[e];                 // inclusive cumsum at this element
    int ci = (int)run;           // int32 truncation (values are positive)
    ci = ci < 0 ? 0 : (ci > H_PIX ? H_PIX : ci);
    dst[tid * EPT + e] = ci;
  }
}

// ---------------------------------------------------------------------------
// Kernel 2: per-pixel histogram fill (with the preserved stale-gap bug) and
// per-(block,channel) deterministic partial sums of (h1-h2)^2.
// ---------------------------------------------------------------------------
__device__ __forceinline__ int upper_bound_le(const int* __restrict__ a,
                                              int n, int p) {
  int lo = 0, hi = n;
  while (lo < hi) {
    int m = (lo + hi) >> 1;
    if (a[m] <= p) lo = m + 1; else hi = m;
  }
  return lo;   // searchsorted(a[:n], p, side='right')
}

__global__ void hist_loss_kernel(const int* __restrict__ cint,
                                 float* __restrict__ partials /* gridDim*3 */) {
  __shared__ float red[THREADS];
  const int tid = threadIdx.x;
  const int p   = blockIdx.x * THREADS + tid;   // pixel index, exact cover

  float h1 = 0.0f, h2 = 0.0f;                   // persistent module state
  for (int i = 0; i < 3; ++i) {
    const int* __restrict__ cA = cint + i * NBINS;        // target cumsum
    const int* __restrict__ cB = cint + (3 + i) * NBINS;  // output cumsum

    const int j1 = upper_bound_le(cA, NBINS - 1, p);
    const int j2 = upper_bound_le(cB, NBINS - 1, p);
    const bool st1 = (p >= cA[NBINS - 2]) && (p < cA[NBINS - 1]);
    const bool st2 = (p >= cB[NBINS - 2]) && (p < cB[NBINS - 1]);
    h1 = st1 ? h1 : (float)j1;   // stale gap keeps previous channel value
    h2 = st2 ? h2 : (float)j2;

    const float d = h1 - h2;
    red[tid] = d * d;
    __syncthreads();
    for (int sft = THREADS / 2; sft > 0; sft >>= 1) {
      if (tid < sft) red[tid] += red[tid + sft];
      __syncthreads();
    }
    if (tid == 0) partials[blockIdx.x * 3 + i] = red[0];
    __syncthreads();
  }
}

// ---------------------------------------------------------------------------
// Kernel 3: deterministic final reduction: loss = sum_i sqrt(sum_b part[b][i])
// ---------------------------------------------------------------------------
__global__ void finalize_kernel(const float* __restrict__ partials, int nblocks,
                                float* __restrict__ out) {
  float loss = 0.0f;
  for (int i = 0; i < 3; ++i) {
    float s = 0.0f;
    for (int b = 0; b < nblocks; ++b) s += partials[b * 3 + i];
    loss += sqrtf(s);
  }
  out[0] = loss;
}

extern "C" void kernel_launch(void* const* d_in, const int* in_sizes, int n_in,
                              void* d_out, int out_size, void* d_ws, size_t ws_size,
                              hipStream_t stream) {
  const float* target = (const float*)d_in[0];   // (3, 8192, 1) f32
  const float* output = (const float*)d_in[1];   // (3, 8192, 1) f32

  int*   cint     = (int*)d_ws;                              // 6*8192 ints
  float* partials = (float*)((char*)d_ws + 6 * NBINS * sizeof(int));

  const int nblocks = H_PIX / THREADS;  // 196, exact

  scan_kernel<<<6, THREADS, 0, stream>>>(target, output, cint);
  hist_loss_kernel<<<nblocks, THREADS, 0, stream>>>(cint, partials);
  finalize_kernel<<<1, 1, 0, stream>>>(partials, nblocks, (float*)d_out);
}